// RGCNConv_50044958933135
// MI455X (gfx1250) — compile-verified
//
#include <hip/hip_runtime.h>

typedef float v2f __attribute__((ext_vector_type(2)));
typedef float v8f __attribute__((ext_vector_type(8)));

#define D 64

// ---------------------------------------------------------------------------
// Zero-fill workspace (grid-stride; deterministic, capture-safe)
// ---------------------------------------------------------------------------
__global__ void zero_f32(float* __restrict__ p, long long n) {
  long long i = (long long)blockIdx.x * blockDim.x + threadIdx.x;
  long long stride = (long long)gridDim.x * blockDim.x;
  for (; i < n; i += stride) p[i] = 0.0f;
}

// ---------------------------------------------------------------------------
// Edge aggregation: segment-sum of source features + degree counts.
// 64 threads cooperate on one edge (one float each of the 64-wide feature).
// ---------------------------------------------------------------------------
__global__ void edge_agg(const float* __restrict__ x,
                         const int* __restrict__ src,
                         const int* __restrict__ dst,
                         float* __restrict__ agg,
                         float* __restrict__ cnt,
                         int E) {
  long long tid = (long long)blockIdx.x * blockDim.x + threadIdx.x;
  int e = (int)(tid >> 6);
  int t = (int)(tid & 63);
  if (e >= E) return;
  int s = src[e];
  int d = dst[e];
  atomicAdd(&agg[(long long)d * D + t], x[(long long)s * D + t]);
  if (t == 0) atomicAdd(&cnt[d], 1.0f);
}

// ---------------------------------------------------------------------------
// One (A rows, 64x64 weight) pair accumulated into 16x64 tile via
// V_WMMA_F32_16X16X4_F32. Per-wave: 4 N-tiles x 16 K-steps = 64 WMMAs.
// A layout (16x4 f32): lane<16 -> row M=lane, K={4k,4k+1}; lane>=16 -> K={4k+2,4k+3}.
// B layout (4x16 f32): mirrored, N = lane&15.
// C/D layout: VGPR r, lane l -> row r + 8*(l>>4), col l&15.
// ---------------------------------------------------------------------------
__device__ __forceinline__ void mma_pair(const float* __restrict__ arow,
                                         float scale,
                                         const float* __restrict__ W,
                                         int n, int kh, v8f acc[4]) {
  v2f a[16];
#pragma unroll
  for (int k = 0; k < 16; ++k) {
    const float* p = arow + 4 * k + 2 * kh;
    v2f v;
    v.x = p[0] * scale;
    v.y = p[1] * scale;
    a[k] = v;
  }
#pragma unroll
  for (int t = 0; t < 4; ++t) {
#pragma unroll
    for (int k = 0; k < 16; ++k) {
      int kk = 4 * k + 2 * kh;
      v2f b;
      b.x = W[kk * D + t * 16 + n];
      b.y = W[(kk + 1) * D + t * 16 + n];
      acc[t] = __builtin_amdgcn_wmma_f32_16x16x4_f32(
          /*neg_a=*/false, a[k], /*neg_b=*/false, b,
          /*c_mod=*/(short)0, acc[t], /*reuse_a=*/false, /*reuse_b=*/false);
    }
  }
}

// ---------------------------------------------------------------------------
// Per-node-type output: out = x@Wroot + b [+ (agg1/max(cnt1,1))@W1]
//                                        [+ (agg2/max(cnt2,1))@W2]
// One wave per 16-row tile; blockDim=256 (8 waves). Wave-uniform guard keeps
// EXEC all-ones around WMMA.
// ---------------------------------------------------------------------------
template <int NREL>
__global__ void hgt_out(const float* __restrict__ x,
                        const float* __restrict__ wroot,
                        const float* __restrict__ bias,
                        const float* __restrict__ agg1,
                        const float* __restrict__ cnt1,
                        const float* __restrict__ w1,
                        const float* __restrict__ agg2,
                        const float* __restrict__ cnt2,
                        const float* __restrict__ w2,
                        float* __restrict__ out,
                        int ntiles) {
  int lane = threadIdx.x & 31;
  int wave = threadIdx.x >> 5;
  int tile = blockIdx.x * (blockDim.x >> 5) + wave;
  if (tile >= ntiles) return;
  int n = lane & 15;
  int kh = lane >> 4;
  long long m = (long long)tile * 16 + n;  // this lane's A row

  v8f acc[4];
#pragma unroll
  for (int t = 0; t < 4; ++t) {
    float bv = bias[t * 16 + n];  // bias broadcast down the column
    v8f c;
#pragma unroll
    for (int r = 0; r < 8; ++r) c[r] = bv;
    acc[t] = c;
  }

  mma_pair(x + m * D, 1.0f, wroot, n, kh, acc);
  if (NREL >= 1) {
    float inv = 1.0f / fmaxf(cnt1[m], 1.0f);
    mma_pair(agg1 + m * D, inv, w1, n, kh, acc);
  }
  if (NREL >= 2) {
    float inv = 1.0f / fmaxf(cnt2[m], 1.0f);
    mma_pair(agg2 + m * D, inv, w2, n, kh, acc);
  }

#pragma unroll
  for (int t = 0; t < 4; ++t) {
#pragma unroll
    for (int r = 0; r < 8; ++r) {
      long long row = (long long)tile * 16 + r + 8 * kh;
      out[row * D + t * 16 + n] = acc[t][r];
    }
  }
}

// ---------------------------------------------------------------------------
// Host launcher
// ---------------------------------------------------------------------------
extern "C" void kernel_launch(void* const* d_in, const int* in_sizes, int n_in,
                              void* d_out, int out_size, void* d_ws, size_t ws_size,
                              hipStream_t stream) {
  const float* x_paper  = (const float*)d_in[0];
  const float* x_author = (const float*)d_in[1];
  const float* x_inst   = (const float*)d_in[2];
  const float* x_field  = (const float*)d_in[3];
  const float* w_root_paper  = (const float*)d_in[4];
  const float* b_root_paper  = (const float*)d_in[5];
  const float* w_root_author = (const float*)d_in[6];
  const float* b_root_author = (const float*)d_in[7];
  const float* w_root_inst   = (const float*)d_in[8];
  const float* b_root_inst   = (const float*)d_in[9];
  const float* w_root_field  = (const float*)d_in[10];
  const float* b_root_field  = (const float*)d_in[11];
  const float* w_cites  = (const float*)d_in[12];
  const float* w_writes = (const float*)d_in[13];
  const float* w_topic  = (const float*)d_in[14];
  const float* w_affil  = (const float*)d_in[15];
  const int* src_cites  = (const int*)d_in[16];
  const int* dst_cites  = (const int*)d_in[17];
  const int* src_writes = (const int*)d_in[18];
  const int* dst_writes = (const int*)d_in[19];
  const int* src_topic  = (const int*)d_in[20];
  const int* dst_topic  = (const int*)d_in[21];
  const int* src_affil  = (const int*)d_in[22];
  const int* dst_affil  = (const int*)d_in[23];

  const long long n_paper  = in_sizes[0] / D;   // 200000
  const long long n_author = in_sizes[1] / D;   // 200000
  const long long n_inst   = in_sizes[2] / D;   // 8000
  const long long n_field  = in_sizes[3] / D;   // 40000
  const int e_cites  = in_sizes[16];
  const int e_writes = in_sizes[18];
  const int e_topic  = in_sizes[20];
  const int e_affil  = in_sizes[22];

  float* out = (float*)d_out;
  float* out_paper  = out;
  float* out_author = out + n_paper * D;
  float* out_inst   = out_author + n_author * D;
  float* out_field  = out_inst + n_inst * D;

  // Workspace layout (floats): 4 agg buffers + 4 count buffers (~116.6 MB)
  float* ws = (float*)d_ws;
  long long off = 0;
  float* aggC = ws + off; off += n_paper * D;
  float* aggW = ws + off; off += n_paper * D;
  float* aggT = ws + off; off += n_field * D;
  float* aggA = ws + off; off += n_inst * D;
  float* cntC = ws + off; off += n_paper;
  float* cntW = ws + off; off += n_paper;
  float* cntT = ws + off; off += n_field;
  float* cntA = ws + off; off += n_inst;
  const long long ws_floats = off;

  // 1) zero agg/count workspace
  zero_f32<<<4096, 256, 0, stream>>>(ws, ws_floats);

  // 2) edge aggregation (segment-sum + degree) per relation
  {
    long long th;
    th = (long long)e_cites * 64;
    edge_agg<<<(unsigned)((th + 255) / 256), 256, 0, stream>>>(
        x_paper, src_cites, dst_cites, aggC, cntC, e_cites);
    th = (long long)e_writes * 64;
    edge_agg<<<(unsigned)((th + 255) / 256), 256, 0, stream>>>(
        x_author, src_writes, dst_writes, aggW, cntW, e_writes);
    th = (long long)e_topic * 64;
    edge_agg<<<(unsigned)((th + 255) / 256), 256, 0, stream>>>(
        x_paper, src_topic, dst_topic, aggT, cntT, e_topic);
    th = (long long)e_affil * 64;
    edge_agg<<<(unsigned)((th + 255) / 256), 256, 0, stream>>>(
        x_author, src_affil, dst_affil, aggA, cntA, e_affil);
  }

  // 3) fused WMMA output kernels (root linear + bias + relation projections)
  const int WAVES = 8;  // 256 threads
  {
    int nt = (int)(n_paper / 16);
    hgt_out<2><<<(nt + WAVES - 1) / WAVES, 32 * WAVES, 0, stream>>>(
        x_paper, w_root_paper, b_root_paper,
        aggC, cntC, w_cites, aggW, cntW, w_writes, out_paper, nt);
  }
  {
    int nt = (int)(n_author / 16);
    hgt_out<0><<<(nt + WAVES - 1) / WAVES, 32 * WAVES, 0, stream>>>(
        x_author, w_root_author, b_root_author,
        nullptr, nullptr, nullptr, nullptr, nullptr, nullptr, out_author, nt);
  }
  {
    int nt = (int)(n_inst / 16);
    hgt_out<1><<<(nt + WAVES - 1) / WAVES, 32 * WAVES, 0, stream>>>(
        x_inst, w_root_inst, b_root_inst,
        aggA, cntA, w_affil, nullptr, nullptr, nullptr, out_inst, nt);
  }
  {
    int nt = (int)(n_field / 16);
    hgt_out<1><<<(nt + WAVES - 1) / WAVES, 32 * WAVES, 0, stream>>>(
        x_field, w_root_field, b_root_field,
        aggT, cntT, w_topic, nullptr, nullptr, nullptr, out_field, nt);
  }
  (void)n_in; (void)out_size; (void)ws_size;
}